// SSIMLoss_3590592659859
// MI455X (gfx1250) — compile-verified
//
#include <hip/hip_runtime.h>
#include <cstdint>

#define AS1 __attribute__((address_space(1)))
#define AS3 __attribute__((address_space(3)))

typedef int v4i __attribute__((ext_vector_type(4)));

#if defined(__has_builtin)
#if __has_builtin(__builtin_amdgcn_global_load_async_to_lds_b128)
#define HAVE_ASYNC_B128 1
#endif
#if __has_builtin(__builtin_amdgcn_s_wait_asynccnt)
#define HAVE_WAIT_ASYNC 1
#endif
#endif

// ---- CDNA5 async global->LDS (ASYNCcnt) helpers -----------------------------
__device__ __forceinline__ void async_tile_load_b128(const float* g, float* l) {
#if defined(HAVE_ASYNC_B128)
  __builtin_amdgcn_global_load_async_to_lds_b128(
      (AS1 v4i*)(v4i*)(uintptr_t)g, (AS3 v4i*)(v4i*)l, 0, 0);
#else
  asm volatile("global_load_async_to_lds_b128 %0, %1, off"
               :
               : "v"((unsigned)(uintptr_t)(AS3 float*)l),
                 "v"((unsigned long long)(uintptr_t)g)
               : "memory");
#endif
}

__device__ __forceinline__ void wait_async_zero() {
#if defined(HAVE_WAIT_ASYNC)
  __builtin_amdgcn_s_wait_asynccnt(0);
#else
  asm volatile("s_wait_asynccnt 0x0" ::: "memory");
#endif
}

// ---- constants --------------------------------------------------------------
#define W       512
#define OUTW    506          // 512 - 7 + 1
#define TILE    32
#define IN      38           // TILE + 6 halo
#define LSTR    40           // LDS row stride (floats), keeps float4 alignment
#define NPIX    49.0f
#define COVN    (49.0f / 48.0f)

// ---- main kernel first in the TU so the disasm snippet shows it -------------
// Tiled SSIM: async-DMA 38x38 X/Y tiles into LDS, separable 7x7 rolling sums,
// wave32 reduce, f64 atomic accumulate.
__global__ __launch_bounds__(256) void ssim_main_kernel(
    const float* __restrict__ X, const float* __restrict__ Y,
    const unsigned* __restrict__ maxb, double* __restrict__ acc) {
  __shared__ __align__(16) float sX[IN * LSTR];
  __shared__ __align__(16) float sY[IN * LSTR];
  __shared__ __align__(16) float csx[TILE * LSTR];
  __shared__ __align__(16) float csy[TILE * LSTR];
  __shared__ __align__(16) float csxx[TILE * LSTR];
  __shared__ __align__(16) float csyy[TILE * LSTR];
  __shared__ __align__(16) float csxy[TILE * LSTR];

  const int tx  = threadIdx.x;            // 0..31 (lane id)
  const int ty  = threadIdx.y;            // 0..7  (wave id)
  const int tid = ty * 32 + tx;
  const int img = blockIdx.z;
  const int ox0 = blockIdx.x * TILE;
  const int oy0 = blockIdx.y * TILE;
  const float* Xi = X + (size_t)img * (W * W);
  const float* Yi = Y + (size_t)img * (W * W);

  // ---- stage 0: async DMA 38x38 tiles of X and Y into LDS (10 float4/row) --
  // Clamped rows/cols only feed windows of out-of-range outputs (discarded).
  for (int idx = tid; idx < 2 * IN * 10; idx += 256) {
    const int a   = (idx >= IN * 10);
    const int rem = a ? idx - IN * 10 : idx;
    const int r   = rem / 10;
    const int k   = rem - r * 10;
    int gy = oy0 + r;     if (gy > W - 1) gy = W - 1;
    int gx = ox0 + 4 * k; if (gx > W - 4) gx = W - 4;
    const float* g = (a ? Yi : Xi) + (size_t)gy * W + gx;
    float*       l = (a ? sY : sX) + r * LSTR + 4 * k;
    async_tile_load_b128(g, l);
  }
  wait_async_zero();        // drain this wave's ASYNCcnt before the barrier
  __syncthreads();

  // ---- stage 1: vertical rolling 7-sums -> 5 column-sum planes -------------
  // Wave ty owns output rows r0..r0+3; needs input rows r0..r0+9.
  const int r0 = ty * 4;
  for (int c = tx; c < IN; c += 32) {
    float qx[10], qy[10];
#pragma unroll
    for (int i = 0; i < 10; ++i) {
      qx[i] = sX[(r0 + i) * LSTR + c];
      qy[i] = fabsf(sY[(r0 + i) * LSTR + c]);
    }
    float ax = 0.f, ay = 0.f, axx = 0.f, ayy = 0.f, axy = 0.f;
#pragma unroll
    for (int i = 0; i < 7; ++i) {
      ax += qx[i]; ay += qy[i];
      axx += qx[i] * qx[i]; ayy += qy[i] * qy[i]; axy += qx[i] * qy[i];
    }
    csx [r0 * LSTR + c] = ax;  csy [r0 * LSTR + c] = ay;
    csxx[r0 * LSTR + c] = axx; csyy[r0 * LSTR + c] = ayy;
    csxy[r0 * LSTR + c] = axy;
#pragma unroll
    for (int j = 1; j < 4; ++j) {
      const float nx = qx[j + 6], ny = qy[j + 6];
      const float px = qx[j - 1], py = qy[j - 1];
      ax  += nx - px;            ay  += ny - py;
      axx += nx * nx - px * px;  ayy += ny * ny - py * py;
      axy += nx * ny - px * py;
      csx [(r0 + j) * LSTR + c] = ax;  csy [(r0 + j) * LSTR + c] = ay;
      csxx[(r0 + j) * LSTR + c] = axx; csyy[(r0 + j) * LSTR + c] = ayy;
      csxy[(r0 + j) * LSTR + c] = axy;
    }
  }
  __syncthreads();

  // ---- stage 2: horizontal rolling 7-sums, SSIM, reduce --------------------
  const int rr = tid >> 3;              // output row in tile, 0..31
  const int c0 = (tid & 7) * 4;         // first of 4 output cols, 0..28

  auto roll4 = [&](const float* __restrict__ cs, float* w) {
    float q[10];
#pragma unroll
    for (int i = 0; i < 10; ++i) q[i] = cs[rr * LSTR + c0 + i];
    float s = q[0] + q[1] + q[2] + q[3] + q[4] + q[5] + q[6];
    w[0] = s; s += q[7] - q[0];
    w[1] = s; s += q[8] - q[1];
    w[2] = s; s += q[9] - q[2];
    w[3] = s;
  };

  float wx[4], wy[4], wxx[4], wyy[4], wxy[4];
  roll4(csx, wx); roll4(csy, wy); roll4(csxx, wxx);
  roll4(csyy, wyy); roll4(csxy, wxy);

  const float mv = __uint_as_float(maxb[img]);
  float C1 = 0.01f * mv; C1 *= C1;
  float C2 = 0.03f * mv; C2 *= C2;
  const float inv = 1.0f / NPIX;

  float local = 0.0f;
  const int oy = oy0 + rr;
#pragma unroll
  for (int j = 0; j < 4; ++j) {
    const int ox = ox0 + c0 + j;
    if (oy < OUTW && ox < OUTW) {
      const float ux  = wx[j] * inv, uy = wy[j] * inv;
      const float vx  = COVN * (wxx[j] * inv - ux * ux);
      const float vy  = COVN * (wyy[j] * inv - uy * uy);
      const float vxy = COVN * (wxy[j] * inv - ux * uy);
      const float A1 = 2.0f * ux * uy + C1;
      const float A2 = 2.0f * vxy + C2;
      const float B1 = ux * ux + uy * uy + C1;
      const float B2 = vx + vy + C2;
      local += 1.0f - (A1 * A2) / (B1 * B2);
    }
  }
  for (int o = 16; o > 0; o >>= 1) local += __shfl_down(local, o, 32);
  if (tx == 0) atomicAdd(acc, (double)local);   // global_atomic_add_f64
}

// ---- kernel 0: zero accumulator + per-image max slots -----------------------
__global__ void ssim_init_kernel(double* acc, unsigned* maxb) {
  if (threadIdx.x == 0) acc[0] = 0.0;
  if (threadIdx.x < 64) maxb[threadIdx.x] = 0u;   // 0u == +0.0f bit pattern
}

// ---- kernel 1: per-image max(|Y|), wave32 reduce + bitwise atomicMax --------
__global__ __launch_bounds__(256) void ssim_absmax_kernel(
    const float* __restrict__ Y, unsigned* __restrict__ maxb) {
  const int img = blockIdx.y;
  const size_t base = (size_t)img * (W * W) + (size_t)blockIdx.x * 32768;
  const float4* p = reinterpret_cast<const float4*>(Y + base);
  float m = 0.0f;
  for (int i = threadIdx.x; i < 8192; i += 256) {
    float4 v = p[i];
    m = fmaxf(m, fmaxf(fmaxf(fabsf(v.x), fabsf(v.y)),
                       fmaxf(fabsf(v.z), fabsf(v.w))));
  }
  for (int o = 16; o > 0; o >>= 1) m = fmaxf(m, __shfl_down(m, o, 32));
  if ((threadIdx.x & 31) == 0)
    atomicMax(maxb + img, __float_as_uint(m));     // all values >= 0
}

// ---- kernel 3: finalize mean ------------------------------------------------
__global__ void ssim_finalize_kernel(const double* acc, float* out) {
  out[0] = (float)(acc[0] * (1.0 / (64.0 * 506.0 * 506.0)));
}

extern "C" void kernel_launch(void* const* d_in, const int* in_sizes, int n_in,
                              void* d_out, int out_size, void* d_ws, size_t ws_size,
                              hipStream_t stream) {
  const float* X = (const float*)d_in[0];
  const float* Y = (const float*)d_in[1];
  float* out = (float*)d_out;
  double*   acc  = (double*)d_ws;                       // 8 bytes
  unsigned* maxb = (unsigned*)((char*)d_ws + 64);       // 64 x u32

  ssim_init_kernel<<<1, 128, 0, stream>>>(acc, maxb);
  ssim_absmax_kernel<<<dim3(8, 64), 256, 0, stream>>>(Y, maxb);
  ssim_main_kernel<<<dim3(16, 16, 64), dim3(32, 8), 0, stream>>>(X, Y, maxb, acc);
  ssim_finalize_kernel<<<1, 1, 0, stream>>>(acc, out);
}